// net_test_57904749085007
// MI455X (gfx1250) — compile-verified
//
#include <hip/hip_runtime.h>

#define NNODES 65536
#define NEDGES 524288
#define FDIM   128
#define ASTRIDE 132   // dword stride for the LDS A tile: conflict-free b64 frag reads

typedef float v2f __attribute__((ext_vector_type(2)));
typedef float v8f __attribute__((ext_vector_type(8)));
typedef int   v4i __attribute__((vector_size(4 * sizeof(int))));

typedef __attribute__((address_space(1))) v4i* g_v4i_ptr;
typedef __attribute__((address_space(3))) v4i* l_v4i_ptr;
typedef __attribute__((address_space(1))) void* g_void_ptr;
typedef __attribute__((address_space(3))) void* l_void_ptr;

#if __has_builtin(__builtin_amdgcn_global_load_async_to_lds_b128)
#define GCN_ASYNC_LDS 1
#else
#define GCN_ASYNC_LDS 0
#endif

__device__ __forceinline__ void copy16_to_lds(float* ldst, const float* gsrc) {
#if GCN_ASYNC_LDS
  g_v4i_ptr gp = (g_v4i_ptr)(g_void_ptr)(void*)gsrc;
  l_v4i_ptr lp = (l_v4i_ptr)(l_void_ptr)(void*)ldst;
  __builtin_amdgcn_global_load_async_to_lds_b128(gp, lp, /*offset=*/0, /*cpol=*/0);
#else
  *(float4*)ldst = *(const float4*)gsrc;
#endif
}

__device__ __forceinline__ void wait_lds_copies() {
#if GCN_ASYNC_LDS
#if __has_builtin(__builtin_amdgcn_s_wait_asynccnt)
  __builtin_amdgcn_s_wait_asynccnt(0);
#else
  asm volatile("s_wait_asynccnt 0x0" ::: "memory");
#endif
#endif
}

__device__ __forceinline__ int lower_bound_i32(const int* __restrict__ a, int n, int key) {
  int lo = 0, hi = n;
  while (lo < hi) {
    int mid = (lo + hi) >> 1;
    if (a[mid] < key) lo = mid + 1; else hi = mid;
  }
  return lo;
}

// out[r,:] = sum over edges with row==r of val * h[col,:]
// One wave32 per output row; 128 features = 32 lanes x float4 (coalesced).
// edge_row is sorted -> per-row edge range via binary search (wave-uniform).
// No atomics, fully deterministic. Gathered matrix (32 MB) is L2-resident.
__global__ __launch_bounds__(256) void spmm_rowwave_kernel(
    const float* __restrict__ h, const int* __restrict__ rows,
    const int* __restrict__ cols, const float* __restrict__ vals,
    float* __restrict__ out) {
  const int wave = threadIdx.x >> 5;
  const int lane = threadIdx.x & 31;
  const int row  = blockIdx.x * 8 + wave;

  const int lo = lower_bound_i32(rows, NEDGES, row);
  const int hi = lower_bound_i32(rows, NEDGES, row + 1);

  float4 acc = make_float4(0.f, 0.f, 0.f, 0.f);
  for (int e = lo; e < hi; ++e) {
    const int   c = cols[e];
    const float v = vals[e];
    const float4 hv = ((const float4*)(h + (size_t)c * FDIM))[lane];
    acc.x = fmaf(v, hv.x, acc.x);
    acc.y = fmaf(v, hv.y, acc.y);
    acc.z = fmaf(v, hv.z, acc.z);
    acc.w = fmaf(v, hv.w, acc.w);
  }
  ((float4*)(out + (size_t)row * FDIM))[lane] = acc;
}

// Tall-skinny GEMM: out[M=65536, NCOL] = (relu?)(A[M,128] @ W[128,NCOL])
// Block = 256 threads = 8 wave32s, 64 output rows per block.
// W (full 128xNCOL) and columns 0-31 of the 64x128 A tile are staged into LDS with
// GLOBAL_LOAD_ASYNC_TO_LDS_B128 (ASYNCcnt), then K is accumulated with
// V_WMMA_F32_16X16X4_F32 (fp32 exact), 32 steps per 16x16 tile.
template <int NCOL, bool RELU>
__global__ __launch_bounds__(256) void gemm_wmma_kernel(
    const float* __restrict__ A, const float* __restrict__ W,
    float* __restrict__ out) {
  constexpr int NT = NCOL / 16;        // N tiles across waves
  constexpr int WM = 8 / NT;           // wave groups stacked over M
  constexpr int MT = 64 / (16 * WM);   // M subtiles per wave

  __shared__ float sW[128 * NCOL];
  __shared__ float sA[64 * ASTRIDE];

  const int row0blk = blockIdx.x * 64;

  // Stage W: linear b128 copies.
  for (int i = threadIdx.x; i < 128 * NCOL / 4; i += 256)
    copy16_to_lds(&sW[i * 4], W + i * 4);
  // Stage A tile: 64 rows x 32 floats = 512 b128 chunks, padded LDS rows
  // (row stride 132 dwords = 528 B, still 16 B aligned).
  for (int i = threadIdx.x; i < 64 * 8; i += 256) {
    const int r = i >> 3, s = i & 7;
    copy16_to_lds(&sA[r * ASTRIDE + s * 4],
                  A + (size_t)(row0blk + r) * 128 + s * 4);
  }
  wait_lds_copies();
  __syncthreads();

  const int wave = threadIdx.x >> 5;
  const int lane = threadIdx.x & 31;
  const int lp   = lane & 15;   // M (for A) / N (for B,D) position
  const int hl   = lane >> 4;   // selects K pair 0/1 vs 2/3, and M+8 for D
  const int nt   = wave % NT;
  const int mgrp = wave / NT;
  const int n0   = nt * 16;
  const int mrow0 = mgrp * (MT * 16);  // wave's first row within the block tile

  v8f acc[MT] = {};

  for (int k = 0; k < 128; k += 4) {
    const int kk = k + 2 * hl;
    // B frag (4x16, KxN): vgpr0 = row K=kk, vgpr1 = row K=kk+1, N = lp
    v2f b;
    b.x = sW[kk * NCOL + n0 + lp];
    b.y = sW[(kk + 1) * NCOL + n0 + lp];
#pragma unroll
    for (int i = 0; i < MT; ++i) {
      // A frag (16x4, MxK): M = lp, K = kk/kk+1 -> contiguous float2 in LDS
      const float2 av =
          *(const float2*)&sA[(mrow0 + i * 16 + lp) * ASTRIDE + kk];
      v2f a;
      a.x = av.x;
      a.y = av.y;
      acc[i] = __builtin_amdgcn_wmma_f32_16x16x4_f32(
          /*neg_a=*/false, a, /*neg_b=*/false, b,
          /*c_mod=*/(short)0, acc[i], /*reuse_a=*/false, /*reuse_b=*/false);
    }
  }

  // D layout: VGPR v -> M = v (lanes 0-15) or v+8 (lanes 16-31), N = lp
#pragma unroll
  for (int i = 0; i < MT; ++i) {
#pragma unroll
    for (int v = 0; v < 8; ++v) {
      float val = acc[i][v];
      if (RELU) val = (val > 0.0f) ? val : 0.0f;
      out[(size_t)(row0blk + mrow0 + i * 16 + v + 8 * hl) * NCOL + n0 + lp] = val;
    }
  }
}

extern "C" void kernel_launch(void* const* d_in, const int* in_sizes, int n_in,
                              void* d_out, int out_size, void* d_ws, size_t ws_size,
                              hipStream_t stream) {
  const float* x     = (const float*)d_in[0];
  const float* w1    = (const float*)d_in[1];
  const float* w2    = (const float*)d_in[2];
  const float* w3    = (const float*)d_in[3];
  const int*   erow  = (const int*)d_in[4];
  const int*   ecol  = (const int*)d_in[5];
  const float* eval_ = (const float*)d_in[6];
  float* out  = (float*)d_out;
  float* bufA = (float*)d_ws;                   // 32 MB
  float* bufB = bufA + (size_t)NNODES * FDIM;   // 32 MB

  dim3 blk(256);
  // h0 = spmm(x)
  spmm_rowwave_kernel<<<NNODES / 8, blk, 0, stream>>>(x, erow, ecol, eval_, bufA);
  // h1 = relu(h0 @ w1)
  gemm_wmma_kernel<128, true><<<NNODES / 64, blk, 0, stream>>>(bufA, w1, bufB);
  // h2 = spmm(h1)
  spmm_rowwave_kernel<<<NNODES / 8, blk, 0, stream>>>(bufB, erow, ecol, eval_, bufA);
  // h3 = relu(h2 @ w2)
  gemm_wmma_kernel<128, true><<<NNODES / 64, blk, 0, stream>>>(bufA, w2, bufB);
  // out = h3 @ w3
  gemm_wmma_kernel<64, false><<<NNODES / 64, blk, 0, stream>>>(bufB, w3, out);
}